// MaskedAutoencoder_13443247636661
// MI455X (gfx1250) — compile-verified
//
#include <hip/hip_runtime.h>
#include <math.h>

// ---------------- problem constants ----------------
#define N_IMG   32
#define L_TOK   50176      // 224*224
#define IN_DIM  150528     // 3*224*224
#define ENC_DIM 1536
#define DEC_DIM 512
#define LEN_KEEP 12544
#define KSPLIT  16
#define KCHUNK  (IN_DIM / KSPLIT)   // 9408 = 64*147
#define KS      64                  // K rows staged per LDS tile
#define NTILE   64                  // N columns per block

typedef float    v2f __attribute__((ext_vector_type(2)));
typedef float    v4f __attribute__((ext_vector_type(4)));
typedef float    v8f __attribute__((ext_vector_type(8)));
typedef unsigned v4u __attribute__((ext_vector_type(4)));
typedef int      v4i __attribute__((ext_vector_type(4)));
typedef int      v8i __attribute__((ext_vector_type(8)));

#if defined(__gfx1250__) && __has_builtin(__builtin_amdgcn_tensor_load_to_lds) && \
    __has_builtin(__builtin_amdgcn_s_wait_tensorcnt)
#define USE_TDM 1
#else
#define USE_TDM 0
#endif

// low 32 bits of a generic pointer to LDS == DS byte address (ISA 10.2)
__device__ __forceinline__ unsigned lds_off(const void* p) {
  return (unsigned)(unsigned long long)(uintptr_t)p;
}

#if USE_TDM
// Issue one TDM 2D tile load: global (rows x cols, row stride = ldb elems, f32) -> LDS.
// D# packing per CDNA5 ISA §8.3-8.6. tensor dims == tile dims -> no OOB path.
// This toolchain exposes the 6-arg builtin: (g0, g1, g2, g3, g4, cpol);
// the trailing int32x8 group is the unused VADDR4 slot -> zeros.
__device__ __forceinline__ void tdm_load_tile(const float* gsrc, unsigned lds_byte,
                                              int rows, int cols, int ldb) {
  unsigned long long ga = (unsigned long long)(uintptr_t)gsrc;
  unsigned long long s0 = (unsigned long long)ldb;   // dim0 stride (elements)
  v4u g0 = {0u, 0u, 0u, 0u};
  g0[0] = 1u;                                        // count=1, user mode, no gather
  g0[1] = lds_byte;                                  // lds_addr
  g0[2] = (unsigned)(ga & 0xFFFFFFFFull);            // global_addr[31:0]
  g0[3] = (unsigned)((ga >> 32) & 0x1FFFFFFull) | (2u << 30); // addr[56:32], type=2
  v8i g1 = {0, 0, 0, 0, 0, 0, 0, 0};
  g1[0] = (int)(2u << 16);                           // wg_mask=0, data_size=2 (4B)
  g1[1] = (int)(((unsigned)cols & 0xFFFFu) << 16);   // tensor_dim0[15:0]
  g1[2] = (int)((((unsigned)cols) >> 16) | (((unsigned)rows & 0xFFFFu) << 16)); // dim0 hi | dim1 lo
  g1[3] = (int)((((unsigned)rows) >> 16) | (((unsigned)cols) << 16));           // dim1 hi | tile_dim0
  g1[4] = (int)(unsigned)rows;                       // tile_dim1 (tile_dim2 = 0)
  g1[5] = (int)(unsigned)(s0 & 0xFFFFFFFFull);       // tensor_dim0_stride[31:0]
  g1[6] = (int)(unsigned)((s0 >> 32) & 0xFFFFull);   // stride hi | dim1_stride lo (=0)
  g1[7] = 0;
  v4i g2 = {0, 0, 0, 0};
  v4i g3 = {0, 0, 0, 0};
  v8i g4 = {0, 0, 0, 0, 0, 0, 0, 0};
  __builtin_amdgcn_tensor_load_to_lds(g0, g1, g2, g3, g4, 0);
}
#endif

// ---------------------------------------------------------------------------
// Block-cooperative GEMM core. block = 256 threads = 8 waves covering a
// 32(M) x 64(N) C tile; weight tiles (KS x NTILE f32) staged in LDS,
// double-buffered via the Tensor Data Mover when available.
// WMMA fragments per ISA 7.12.2 (V_WMMA_F32_16X16X4_F32, wave32, full EXEC).
// ---------------------------------------------------------------------------
__device__ __forceinline__ v8f gemm_core(const float* __restrict__ A, int lda,
                                         const float* __restrict__ B, int ldb,
                                         float* sB,            // __shared__, 2*KS*NTILE floats
                                         int n_blk, int k_begin, int k_end, v8f acc) {
  const int tid   = threadIdx.x;
  const int lane  = tid & 31;
  const int wave  = tid >> 5;
  const int half  = lane >> 4;                    // K-pair half for A/B fragments
  const int m_base = (wave & 1) * 16;
  const int n_sub  = (wave >> 1) * 16;            // wave's columns inside the LDS tile
  const int mrow  = m_base + (lane & 15);
  const int c     = n_sub + (lane & 15);          // LDS-tile column this lane owns
  const float* __restrict__ Arow = A + (size_t)mrow * lda;
  const int ntiles = (k_end - k_begin) / KS;

#define COMPUTE_TILE(bt, k0)                                                     \
  _Pragma("unroll")                                                              \
  for (int ks = 0; ks < KS; ks += 4) {                                           \
    const int kk = ks + (half << 1);                                             \
    v2f afrag, bfrag;                                                            \
    afrag.x = Arow[(k0) + kk];                                                   \
    afrag.y = Arow[(k0) + kk + 1];                                               \
    bfrag.x = (bt)[kk * NTILE + c];                                              \
    bfrag.y = (bt)[(kk + 1) * NTILE + c];                                        \
    acc = __builtin_amdgcn_wmma_f32_16x16x4_f32(false, afrag, false, bfrag,      \
                                                (short)0, acc, false, false);    \
  }

#if USE_TDM
  if (tid < 32)  // wave 0 drives the DMA (wave-uniform branch; EXEC stays full)
    tdm_load_tile(B + (size_t)k_begin * ldb + n_blk, lds_off(sB), KS, NTILE, ldb);
  for (int s = 0; s < ntiles; s++) {
    const int k0 = k_begin + s * KS;
    if (tid < 32) __builtin_amdgcn_s_wait_tensorcnt(0);   // tile s landed in LDS
    __syncthreads();                                       // publish; prior compute done
    if (tid < 32 && (s + 1) < ntiles)                      // prefetch tile s+1 via TDM
      tdm_load_tile(B + (size_t)(k0 + KS) * ldb + n_blk,
                    lds_off(sB + ((s + 1) & 1) * (KS * NTILE)), KS, NTILE, ldb);
    const float* bt = sB + (s & 1) * (KS * NTILE);
    COMPUTE_TILE(bt, k0)
  }
#else
  // fallback: synchronous b128 staging through registers
  const int grp  = tid & 15;        // column group of 4 floats
  const int row0 = tid >> 4;        // rows row0, +16, +32, +48
  for (int s = 0; s < ntiles; s++) {
    const int k0 = k_begin + s * KS;
    const float* gp = B + (size_t)(k0 + row0) * ldb + n_blk + grp * 4;
    v4f r0 = *(const v4f*)gp;
    v4f r1 = *(const v4f*)(gp + (size_t)16 * ldb);
    v4f r2 = *(const v4f*)(gp + (size_t)32 * ldb);
    v4f r3 = *(const v4f*)(gp + (size_t)48 * ldb);
    __syncthreads();                // previous tile fully consumed
    *(v4f*)&sB[(row0 +  0) * NTILE + grp * 4] = r0;
    *(v4f*)&sB[(row0 + 16) * NTILE + grp * 4] = r1;
    *(v4f*)&sB[(row0 + 32) * NTILE + grp * 4] = r2;
    *(v4f*)&sB[(row0 + 48) * NTILE + grp * 4] = r3;
    __syncthreads();
    COMPUTE_TILE(sB, k0)
  }
#endif
#undef COMPUTE_TILE
  return acc;
}

// map float -> monotonic unsigned key (total order matching float compare)
__device__ __forceinline__ unsigned keymap(float f) {
  unsigned u = __float_as_uint(f);
  return (u & 0x80000000u) ? ~u : (u | 0x80000000u);
}

// ---------------------------------------------------------------------------
// Kernel 1: per-row exact k-th order statistic via 3-level radix histogram,
// then write keep-mask (1.0 = keep). Stable tie handling matches argsort.
// ---------------------------------------------------------------------------
__global__ void select_mask_kernel(const float* __restrict__ noise,
                                   float* __restrict__ mask) {
  __shared__ unsigned hist[2048];
  __shared__ unsigned sh[8];
  const int row = blockIdx.x;
  const int t   = threadIdx.x;
  const float* nr = noise + (size_t)row * L_TOK;

  for (int i = t; i < 2048; i += 256) hist[i] = 0;
  __syncthreads();
  for (int i = t; i < L_TOK; i += 256)
    atomicAdd(&hist[keymap(nr[i]) >> 21], 1u);
  __syncthreads();
  if (t == 0) {
    unsigned cum = 0, b = 0;
    for (b = 0; b < 2048; b++) { unsigned h = hist[b]; if (cum + h >= (unsigned)LEN_KEEP) break; cum += h; }
    sh[0] = b; sh[3] = cum;
  }
  __syncthreads();
  const unsigned b0 = sh[0], less1 = sh[3];

  for (int i = t; i < 2048; i += 256) hist[i] = 0;
  __syncthreads();
  for (int i = t; i < L_TOK; i += 256) {
    unsigned key = keymap(nr[i]);
    if ((key >> 21) == b0) atomicAdd(&hist[(key >> 10) & 0x7FFu], 1u);
  }
  __syncthreads();
  if (t == 0) {
    unsigned k = (unsigned)LEN_KEEP - less1, cum = 0, b = 0;
    for (b = 0; b < 2048; b++) { unsigned h = hist[b]; if (cum + h >= k) break; cum += h; }
    sh[1] = b; sh[4] = cum;
  }
  __syncthreads();
  const unsigned b1 = sh[1], less2 = less1 + sh[4];
  const unsigned top22 = (b0 << 11) | b1;

  for (int i = t; i < 1024; i += 256) hist[i] = 0;
  __syncthreads();
  for (int i = t; i < L_TOK; i += 256) {
    unsigned key = keymap(nr[i]);
    if ((key >> 10) == top22) atomicAdd(&hist[key & 0x3FFu], 1u);
  }
  __syncthreads();
  if (t == 0) {
    unsigned k = (unsigned)LEN_KEEP - less2, cum = 0, b = 0;
    for (b = 0; b < 1024; b++) { unsigned h = hist[b]; if (cum + h >= k) break; cum += h; }
    sh[2] = b; sh[5] = cum; sh[6] = hist[b];
  }
  __syncthreads();
  const unsigned T        = (top22 << 10) | sh[2];
  const unsigned cnt_eq   = sh[6];
  const unsigned quota    = (unsigned)LEN_KEEP - (less2 + sh[5]);

  float* mr = mask + (size_t)row * L_TOK;
  for (int i = t; i < L_TOK; i += 256) {
    unsigned key = keymap(nr[i]);
    float m = 0.0f;
    if (key < T) m = 1.0f;
    else if (key == T) {
      if (quota >= cnt_eq) m = 1.0f;
      else {                       // rare tie straddling the threshold: stable rank
        unsigned rnk = 0;
        for (int j = 0; j < i; j++) if (keymap(nr[j]) == T) rnk++;
        m = (rnk < quota) ? 1.0f : 0.0f;
      }
    }
    mr[i] = m;
  }
}

// Kernel 2: masked activation  A[n, c*L+p] = imgs[n, c*L+p] * mask[n, p]
__global__ void apply_mask_kernel(const float* __restrict__ imgs,
                                  const float* __restrict__ mask,
                                  float* __restrict__ A) {
  size_t idx = (size_t)blockIdx.x * 256 + threadIdx.x;  // exact: 32*150528 threads
  int n = (int)(idx / IN_DIM);
  int i = (int)(idx % IN_DIM);
  int p = i % L_TOK;
  A[idx] = imgs[idx] * mask[(size_t)n * L_TOK + p];
}

// Kernel 3: encoder GEMM, split-K partials P[s][32][1536] (streams 925 MB of enc_w)
__global__ void enc_gemm_kernel(const float* __restrict__ A,
                                const float* __restrict__ W,
                                float* __restrict__ P) {
  __shared__ float sB[2 * KS * NTILE];
  const int wave   = threadIdx.x >> 5;
  const int m_base = (wave & 1) * 16;
  const int n_blk  = blockIdx.x * NTILE;
  const int s      = blockIdx.y;
  v8f acc = {};
  acc = gemm_core(A, IN_DIM, W, ENC_DIM, sB, n_blk, s * KCHUNK, s * KCHUNK + KCHUNK, acc);
  const int lane = threadIdx.x & 31;
  const int col  = n_blk + (wave >> 1) * 16 + (lane & 15);
  float* Ps = P + (size_t)s * (N_IMG * ENC_DIM);
#pragma unroll
  for (int r = 0; r < 8; r++) {
    int row = m_base + r + ((lane >> 4) << 3);
    Ps[(size_t)row * ENC_DIM + col] = acc[r];
  }
}

// Kernel 4: latent = bias + sum_s P[s]  (fixed order -> deterministic)
__global__ void reduce_latent_kernel(const float* __restrict__ P,
                                     const float* __restrict__ bias,
                                     float* __restrict__ latent) {
  int idx = blockIdx.x * 256 + threadIdx.x;       // exact: 32*1536 threads
  int n = idx % ENC_DIM;
  float s = bias[n];
#pragma unroll
  for (int i = 0; i < KSPLIT; i++) s += P[(size_t)i * (N_IMG * ENC_DIM) + idx];
  latent[idx] = s;
}

// Kernel 5: decoder layer  Y = gelu_exact(X @ W + b), M = 32
__global__ void dec_gemm_gelu_kernel(const float* __restrict__ X,
                                     const float* __restrict__ W,
                                     const float* __restrict__ b,
                                     float* __restrict__ Y,
                                     int Kdim, int Ndim) {
  __shared__ float sB[2 * KS * NTILE];
  const int wave   = threadIdx.x >> 5;
  const int m_base = (wave & 1) * 16;
  const int n_blk  = blockIdx.x * NTILE;
  v8f acc = {};
  acc = gemm_core(X, Kdim, W, Ndim, sB, n_blk, 0, Kdim, acc);
  const int lane = threadIdx.x & 31;
  const int col  = n_blk + (wave >> 1) * 16 + (lane & 15);
  const float bc = b[col];
#pragma unroll
  for (int r = 0; r < 8; r++) {
    int row = m_base + r + ((lane >> 4) << 3);
    float v = acc[r] + bc;
    float g = v * 0.5f * (1.0f + erff(v * 0.70710678118654752f));  // exact GELU
    Y[(size_t)row * Ndim + col] = g;
  }
}

// Kernel 6: pred = H @ W3 + b3, store pred, block-local SSE (streams 308 MB of w3)
__global__ void out_gemm_loss_kernel(const float* __restrict__ H,
                                     const float* __restrict__ W3,
                                     const float* __restrict__ b3,
                                     const float* __restrict__ imgs,
                                     float* __restrict__ out,      // [0]=loss, [1..]=pred
                                     float* __restrict__ loss_part) {
  __shared__ float sB[2 * KS * NTILE];
  __shared__ float red[256];
  const int wave   = threadIdx.x >> 5;
  const int m_base = (wave & 1) * 16;
  const int n_blk  = blockIdx.x * NTILE;
  v8f acc = {};
  acc = gemm_core(H, DEC_DIM, W3, IN_DIM, sB, n_blk, 0, DEC_DIM, acc);
  const int lane = threadIdx.x & 31;
  const int col  = n_blk + (wave >> 1) * 16 + (lane & 15);
  const float bc = b3[col];
  float* pred = out + 1;
  float lsum = 0.0f;
#pragma unroll
  for (int r = 0; r < 8; r++) {
    int row = m_base + r + ((lane >> 4) << 3);
    size_t idx = (size_t)row * IN_DIM + col;
    float v = acc[r] + bc;
    pred[idx] = v;
    float e = v - imgs[idx];
    lsum += e * e;
  }
  red[threadIdx.x] = lsum;
  __syncthreads();
  for (int s = 128; s > 0; s >>= 1) {
    if (threadIdx.x < s) red[threadIdx.x] += red[threadIdx.x + s];
    __syncthreads();
  }
  if (threadIdx.x == 0) loss_part[blockIdx.x] = red[0];
}

// Kernel 7: final loss reduction (single block, fixed order)
__global__ void loss_reduce_kernel(const float* __restrict__ part,
                                   float* __restrict__ out, int nparts) {
  __shared__ float red[256];
  float s = 0.0f;
  for (int i = threadIdx.x; i < nparts; i += 256) s += part[i];
  red[threadIdx.x] = s;
  __syncthreads();
  for (int k = 128; k > 0; k >>= 1) {
    if (threadIdx.x < k) red[threadIdx.x] += red[threadIdx.x + k];
    __syncthreads();
  }
  if (threadIdx.x == 0)
    out[0] = red[0] / (float)((double)N_IMG * (double)IN_DIM);
}

// ---------------------------------------------------------------------------
extern "C" void kernel_launch(void* const* d_in, const int* in_sizes, int n_in,
                              void* d_out, int out_size, void* d_ws, size_t ws_size,
                              hipStream_t stream) {
  const float* imgs  = (const float*)d_in[0];
  const float* noise = (const float*)d_in[1];
  const float* enc_w = (const float*)d_in[2];
  const float* enc_b = (const float*)d_in[3];
  const float* w1    = (const float*)d_in[4];
  const float* b1    = (const float*)d_in[5];
  const float* w2    = (const float*)d_in[6];
  const float* b2    = (const float*)d_in[7];
  const float* w3    = (const float*)d_in[8];
  const float* b3    = (const float*)d_in[9];
  float* out = (float*)d_out;

  // workspace layout (floats)
  float* ws    = (float*)d_ws;
  float* mask  = ws;                                      // 32*50176
  float* Am    = mask + (size_t)N_IMG * L_TOK;            // 32*150528
  float* P     = Am   + (size_t)N_IMG * IN_DIM;           // 16*32*1536
  float* lat   = P    + (size_t)KSPLIT * N_IMG * ENC_DIM; // 32*1536
  float* hdn1  = lat  + (size_t)N_IMG * ENC_DIM;          // 32*512
  float* hdn2  = hdn1 + (size_t)N_IMG * DEC_DIM;          // 32*512
  float* lpart = hdn2 + (size_t)N_IMG * DEC_DIM;          // 2352

  select_mask_kernel<<<N_IMG, 256, 0, stream>>>(noise, mask);
  apply_mask_kernel<<<(N_IMG * IN_DIM) / 256, 256, 0, stream>>>(imgs, mask, Am);
  enc_gemm_kernel<<<dim3(ENC_DIM / NTILE, KSPLIT), 256, 0, stream>>>(Am, enc_w, P);
  reduce_latent_kernel<<<(N_IMG * ENC_DIM) / 256, 256, 0, stream>>>(P, enc_b, lat);
  dec_gemm_gelu_kernel<<<DEC_DIM / NTILE, 256, 0, stream>>>(lat,  w1, b1, hdn1, ENC_DIM, DEC_DIM);
  dec_gemm_gelu_kernel<<<DEC_DIM / NTILE, 256, 0, stream>>>(hdn1, w2, b2, hdn2, DEC_DIM, DEC_DIM);
  out_gemm_loss_kernel<<<IN_DIM / NTILE, 256, 0, stream>>>(hdn2, w3, b3, imgs, out, lpart);
  loss_reduce_kernel<<<1, 256, 0, stream>>>(lpart, out, IN_DIM / NTILE);
}